// SlidingAttnScoreCache_3564822855690
// MI455X (gfx1250) — compile-verified
//
#include <hip/hip_runtime.h>
#include <stdint.h>

// Problem constants (from reference): B=8, H=8, S=1024, D=64
#define SEQ 1024
#define HDIM 64
#define PLANES 64  // B*H

// ---- CDNA5 async global<->LDS helpers (gfx1250) -------------------------
// Low 32 bits of a generic pointer to __shared__ data are the LDS byte
// offset (ISA 10.2 aperture rules).
__device__ __forceinline__ uint32_t lds_addr32(const void* p) {
    return (uint32_t)(uintptr_t)p;
}

__device__ __forceinline__ void async_load_b128_nt(uint32_t lds_off, const void* gaddr) {
    // Streaming read: non-temporal, don't pollute L2 with one-shot data.
    asm volatile("global_load_async_to_lds_b128 %0, %1, off th:TH_LOAD_NT"
                 :: "v"(lds_off), "v"(gaddr) : "memory");
}

__device__ __forceinline__ void async_load_b128(uint32_t lds_off, const void* gaddr) {
    asm volatile("global_load_async_to_lds_b128 %0, %1, off"
                 :: "v"(lds_off), "v"(gaddr) : "memory");
}

__device__ __forceinline__ void async_store_b128_nt(void* gaddr, uint32_t lds_off) {
    asm volatile("global_store_async_from_lds_b128 %0, %1, off th:TH_STORE_NT"
                 :: "v"(gaddr), "v"(lds_off) : "memory");
}

__device__ __forceinline__ void wait_async0() {
    asm volatile("s_wait_asynccnt 0x0" ::: "memory");
}

__device__ __forceinline__ void wait_ds0() {
    asm volatile("s_wait_dscnt 0x0" ::: "memory");
}

// ---- Kernel 1: q/k/v cache row-shift + append ---------------------------
// out[b,h,s,:] = cache[b,h, cap ? s+1 : s, :]  (s != csl)
// out[b,h,csl,:] = new vector
// grid = (1024, 3): each thread bounces 4 x float4 (64B) through LDS with
// 4 async loads in flight before one wait. grid.y selects q/k/v.
__global__ __launch_bounds__(256) void proj_shift_kernel(
    const float* __restrict__ qc, const float* __restrict__ kc,
    const float* __restrict__ vc,
    const float* __restrict__ qn, const float* __restrict__ kn,
    const float* __restrict__ vn,
    const int* __restrict__ cslp,
    float* __restrict__ out /* 3 * 4M floats, q|k|v */) {
    __shared__ __align__(16) float4 stage[4][256];

    const int t     = threadIdx.x;
    const int which = blockIdx.y;                        // 0=q, 1=k, 2=v
    const long base = (long)blockIdx.x * 1024;           // 4 chunks of 256 float4

    const int  csl  = cslp[0];
    const bool cap  = (csl >= SEQ - 1);
    const int  cslc = cap ? (SEQ - 1) : csl;

    const float* cache = (which == 0) ? qc : (which == 1) ? kc : vc;
    const float* nv    = (which == 0) ? qn : (which == 1) ? kn : vn;

    // Issue 4 coalesced async loads (ASYNCcnt -> 4), then wait once.
#pragma unroll
    for (int k = 0; k < 4; ++k) {
        const long i     = base + (long)k * 256 + t;     // float4 idx in one cache
        const int  d4    = (int)(i & 15);                // D/4 = 16 float4 per row
        const int  s     = (int)((i >> 4) & (SEQ - 1));
        const int  plane = (int)(i >> 14);               // (b*H + h)

        const float4* src;
        if (s == cslc) {
            src = (const float4*)(nv + (size_t)plane * HDIM) + d4;   // tiny, reused? no: RT ok
            async_load_b128(lds_addr32(&stage[k][t]), src);
        } else {
            const int ss = cap ? (s + 1) : s;
            src = (const float4*)(cache + ((size_t)plane * SEQ + ss) * HDIM) + d4;
            async_load_b128_nt(lds_addr32(&stage[k][t]), src);       // streamed once
        }
    }
    wait_async0();

    float4* outc = (float4*)(out + (size_t)which * (size_t)PLANES * SEQ * HDIM);
#pragma unroll
    for (int k = 0; k < 4; ++k) {
        const long i = base + (long)k * 256 + t;
        async_store_b128_nt(outc + i, lds_addr32(&stage[k][t]));     // written once
    }
    wait_async0();
}

// ---- Kernel 2: attn score double-shift + q_t row + k_t col --------------
// At capacity: out[r,c] = in[min(r+2,1023), min(c+2,1023)]; row 1023 <- q_t;
// col 1023 <- k_t (col write wins at [1023,1023]).
// Below capacity: out = in, row csl <- q_t, col csl <- k_t.
// One workgroup (256 thr) per 4KB output row; the row is staged through LDS
// so the 8-byte column shift never produces a misaligned 128-bit access.
__global__ __launch_bounds__(256) void attn_shift_kernel(
    const float* __restrict__ attn_in, const float* __restrict__ q_t,
    const float* __restrict__ k_t, const int* __restrict__ cslp,
    float* __restrict__ out_attn) {
    __shared__ __align__(16) float ldsin[SEQ];
    __shared__ __align__(16) float ldsout[SEQ];

    const int t     = threadIdx.x;
    const int r     = blockIdx.x;
    const int plane = blockIdx.y;

    const int  csl   = cslp[0];
    const bool cap   = (csl >= SEQ - 1);
    const int  cslc  = cap ? (SEQ - 1) : csl;
    const int  shift = (r == cslc) ? 0 : (cap ? 2 : 0);

    const float* srcrow;
    bool stream_nt;
    if (r == cslc) {
        srcrow   = q_t + (size_t)plane * SEQ;            // q_t[b,h,0,:] (tiny)
        stream_nt = false;
    } else {
        int sr = cap ? (r + 2) : r;
        if (sr > SEQ - 1) sr = SEQ - 1;
        srcrow   = attn_in + ((size_t)plane * SEQ + sr) * SEQ;  // 256MB stream
        stream_nt = true;
    }

    // Stage aligned source row into LDS: 256 lanes x 16B = 4KB row.
    const uint32_t li = lds_addr32(&ldsin[4 * t]);
    if (stream_nt) async_load_b128_nt(li, (const float4*)srcrow + t);
    else           async_load_b128(li, (const float4*)srcrow + t);
    wait_async0();
    __syncthreads();                                     // cross-wave LDS visibility

    const float ktv = k_t[(size_t)plane * SEQ + r];      // k_t[b,h,r,0] (tiny, RT)

#pragma unroll
    for (int j = 0; j < 4; ++j) {
        const int c = 4 * t + j;
        float v;
        if (c == cslc) {
            v = ktv;                                     // column write wins
        } else {
            int cs = c + shift;
            if (cs > SEQ - 1) cs = SEQ - 1;              // clamp (duplicated edge)
            v = ldsin[cs];
        }
        ldsout[c] = v;                                   // own 16B slot
    }

    wait_ds0();                                          // own LDS writes committed
    const uint32_t lo = lds_addr32(&ldsout[4 * t]);
    float4* gdst = (float4*)(out_attn + ((size_t)plane * SEQ + r) * SEQ) + t;
    async_store_b128_nt(gdst, lo);                       // written once, never re-read
    wait_async0();
}

// ---- Host launch --------------------------------------------------------
extern "C" void kernel_launch(void* const* d_in, const int* in_sizes, int n_in,
                              void* d_out, int out_size, void* d_ws, size_t ws_size,
                              hipStream_t stream) {
    const float* q_cache = (const float*)d_in[0];
    const float* k_cache = (const float*)d_in[1];
    const float* v_cache = (const float*)d_in[2];
    const float* attn_in = (const float*)d_in[3];
    const float* q_new   = (const float*)d_in[4];
    const float* k_new   = (const float*)d_in[5];
    const float* v_new   = (const float*)d_in[6];
    const float* q_t     = (const float*)d_in[7];
    const float* k_t     = (const float*)d_in[8];
    const int*   cslp    = (const int*)d_in[9];

    float* out = (float*)d_out;
    // Output layout: q_cache | k_cache | v_cache | attn  (flat, return order)
    float* out_qkv  = out;
    float* out_attn = out + (size_t)3 * PLANES * SEQ * HDIM;  // 12,582,912 floats in

    // 1M float4 per cache / (256 thr * 4 per thr) = 1024 blocks; grid.y = cache
    dim3 g1(1024, 3, 1);
    proj_shift_kernel<<<g1, 256, 0, stream>>>(q_cache, k_cache, v_cache,
                                              q_new, k_new, v_new, cslp, out_qkv);

    // one workgroup per (plane, row)
    dim3 g2(SEQ, PLANES, 1);
    attn_shift_kernel<<<g2, 256, 0, stream>>>(attn_in, q_t, k_t, cslp, out_attn);
}